// ImprovedGCN_4647154614788
// MI455X (gfx1250) — compile-verified
//
#include <hip/hip_runtime.h>

typedef float v2f __attribute__((ext_vector_type(2)));
typedef float v8f __attribute__((ext_vector_type(8)));

#define GCN_EPS 1e-5f

// ---------------- utility kernels ----------------

__global__ void __launch_bounds__(256) k_zero(float* __restrict__ p, size_t n) {
  size_t i = (size_t)blockIdx.x * blockDim.x + threadIdx.x;
  size_t stride = (size_t)gridDim.x * blockDim.x;
  for (; i < n; i += stride) p[i] = 0.0f;
}

__global__ void __launch_bounds__(256) k_deg_init(float* __restrict__ deg, int n) {
  int i = blockIdx.x * blockDim.x + threadIdx.x;
  if (i < n) deg[i] = 1.0f;  // self-loop contribution
}

__global__ void __launch_bounds__(256) k_deg_accum(const int* __restrict__ dst,
                                                   float* __restrict__ deg, int e) {
  int i = blockIdx.x * blockDim.x + threadIdx.x;
  if (i < e) atomicAdd(&deg[dst[i]], 1.0f);
}

__global__ void __launch_bounds__(256) k_dinv(float* __restrict__ deg, int n) {
  int i = blockIdx.x * blockDim.x + threadIdx.x;
  if (i < n) deg[i] = rsqrtf(deg[i]);  // deg >= 1 always (self-loops)
}

// ---------------- GEMM: out[N,64] = in[N,64] @ W[64,64] ----------------
// One wave per 16x16 output tile; K consumed 4 at a time via
// V_WMMA_F32_16X16X4_F32 (fp32 in, fp32 accumulate).
//
// fp32 WMMA lane layouts (ISA 7.12.2):
//  A (16x4):  lanes 0-15 -> M=lane, VGPR0=K0 VGPR1=K1; lanes 16-31 -> K2,K3
//  B (4x16):  lanes 0-15 -> N=lane, VGPR0=row K0 VGPR1=row K1; lanes 16-31 -> K2,K3
//  C/D(16x16): VGPR r: lanes 0-15 -> M=r, N=lane; lanes 16-31 -> M=8+r, N=lane-16

__global__ void __launch_bounds__(256) k_gemm_n64(const float* __restrict__ A,
                                                  const float* __restrict__ W,
                                                  float* __restrict__ Out,
                                                  int n) {
  const int wave = threadIdx.x >> 5;         // 0..7 (8 waves / block)
  const int lane = threadIdx.x & 31;
  const int rowTile = blockIdx.x * 2 + (wave >> 2);  // 16-row tile index
  const int colTile = wave & 3;                       // 16-col tile index
  const int m0 = rowTile * 16;
  const int n0 = colTile * 16;
  if (m0 >= n) return;  // wave-uniform guard (EXEC stays all-ones for WMMA)

  const int  lhalf = lane & 15;
  const bool hi    = (lane >= 16);
  const int  m     = m0 + lhalf;   // A row for this lane
  const int  nn    = n0 + lhalf;   // B/D column for this lane

  const float* __restrict__ arow = A + (size_t)m * 64;

#if __has_builtin(__builtin_amdgcn_wmma_f32_16x16x4_f32)
  v8f c = {};
#pragma unroll
  for (int k0 = 0; k0 < 64; k0 += 4) {
    const int ka = k0 + (hi ? 2 : 0);
    v2f a = *(const v2f*)(arow + ka);                       // global_load_b64
    v2f b = { W[(size_t)ka * 64 + nn], W[(size_t)(ka + 1) * 64 + nn] };
    c = __builtin_amdgcn_wmma_f32_16x16x4_f32(
        /*neg_a=*/false, a, /*neg_b=*/false, b,
        /*c_mod=*/(short)0, c, /*reuse_a=*/false, /*reuse_b=*/false);
  }
  const int mBase = m0 + (hi ? 8 : 0);
#pragma unroll
  for (int r = 0; r < 8; ++r)
    Out[(size_t)(mBase + r) * 64 + nn] = c[r];
#else
  // Fallback (should not be taken on gfx1250): scalar dot products.
  float acc[8];
#pragma unroll
  for (int r = 0; r < 8; ++r) acc[r] = 0.0f;
  const int mBase = m0 + (hi ? 8 : 0);
#pragma unroll
  for (int r = 0; r < 8; ++r) {
    const float* row = A + (size_t)(mBase + r) * 64;
    float s = 0.0f;
    for (int k = 0; k < 64; ++k) s = fmaf(row[k], W[(size_t)k * 64 + nn], s);
    acc[r] = s;
  }
#pragma unroll
  for (int r = 0; r < 8; ++r)
    Out[(size_t)(mBase + r) * 64 + nn] = acc[r];
#endif
}

// ---------------- edge scatter: agg[dst] += hw[src] * dinv[src]*dinv[dst] ---
// One wave per edge (incl. N trailing self-loops); lane f handles features
// f and f+32 -> fully coalesced 128B gathers and global_atomic_add_f32.

__global__ void __launch_bounds__(256) k_scatter(const int* __restrict__ src,
                                                 const int* __restrict__ dst,
                                                 const float* __restrict__ dinv,
                                                 const float* __restrict__ hw,
                                                 float* __restrict__ agg,
                                                 int E, int N) {
  const int gw   = (int)(((size_t)blockIdx.x * blockDim.x + threadIdx.x) >> 5);
  const int lane = threadIdx.x & 31;
  const int total = E + N;
  if (gw >= total) return;  // wave-uniform

  int s, d;
  if (gw < E) { s = src[gw]; d = dst[gw]; }
  else        { s = gw - E;  d = s; }     // self-loop

  const float w = dinv[s] * dinv[d];
  const float* __restrict__ row  = hw  + (size_t)s * 64;
  float* __restrict__       orow = agg + (size_t)d * 64;

  const float v0 = row[lane]      * w;
  const float v1 = row[lane + 32] * w;
  atomicAdd(&orow[lane],      v0);
  atomicAdd(&orow[lane + 32], v1);
}

// ---------------- BatchNorm (training stats) + bias + ReLU ----------------

__global__ void __launch_bounds__(256) k_bn_stats(const float* __restrict__ agg,
                                                  const float* __restrict__ bias,
                                                  float* __restrict__ stats,  // [128]
                                                  int n) {
  __shared__ float s_sum[64];
  __shared__ float s_sq[64];
  const int tid = threadIdx.x;  // 256
  if (tid < 64) { s_sum[tid] = 0.0f; s_sq[tid] = 0.0f; }
  __syncthreads();

  const int f = tid & 63;
  const float b = bias[f];
  float lsum = 0.0f, lsq = 0.0f;
  const int rowStride = gridDim.x * 4;
  for (int r = blockIdx.x * 4 + (tid >> 6); r < n; r += rowStride) {
    const float v = agg[(size_t)r * 64 + f] + b;
    lsum += v;
    lsq  += v * v;
  }
  atomicAdd(&s_sum[f], lsum);
  atomicAdd(&s_sq[f], lsq);
  __syncthreads();
  if (tid < 64) {
    atomicAdd(&stats[tid],      s_sum[tid]);
    atomicAdd(&stats[64 + tid], s_sq[tid]);
  }
}

__global__ void k_bn_finalize(const float* __restrict__ stats,
                              const float* __restrict__ gamma,
                              const float* __restrict__ beta,
                              float* __restrict__ coef,  // [128]: scale, shift
                              float invN) {
  const int f = threadIdx.x;  // 64 threads
  const float m = stats[f] * invN;
  const float v = stats[64 + f] * invN - m * m;       // biased variance
  const float s = gamma[f] * rsqrtf(v + GCN_EPS);
  coef[f]      = s;
  coef[64 + f] = beta[f] - m * s;
}

__global__ void __launch_bounds__(256) k_bn_apply_relu(const float* __restrict__ agg,
                                                       const float* __restrict__ bias,
                                                       const float* __restrict__ coef,
                                                       float* __restrict__ out,
                                                       size_t n64) {
  const size_t i = (size_t)blockIdx.x * blockDim.x + threadIdx.x;
  if (i < n64) {
    const int f = (int)(i & 63);
    const float v = (agg[i] + bias[f]) * coef[f] + coef[64 + f];
    out[i] = v > 0.0f ? v : 0.0f;
  }
}

__global__ void __launch_bounds__(256) k_final_add(const float* __restrict__ agg,
                                                   const float* __restrict__ b3,
                                                   const float* __restrict__ x,
                                                   float* __restrict__ out,
                                                   size_t n64) {
  const size_t i = (size_t)blockIdx.x * blockDim.x + threadIdx.x;
  if (i < n64) out[i] = agg[i] + b3[(int)(i & 63)] + x[i];
}

// ---------------- host orchestration ----------------

extern "C" void kernel_launch(void* const* d_in, const int* in_sizes, int n_in,
                              void* d_out, int out_size, void* d_ws, size_t ws_size,
                              hipStream_t stream) {
  const float* x   = (const float*)d_in[0];
  const int*   ei  = (const int*)d_in[1];
  const float* W1  = (const float*)d_in[2];
  const float* b1  = (const float*)d_in[3];
  const float* g1  = (const float*)d_in[4];
  const float* be1 = (const float*)d_in[5];
  const float* W2  = (const float*)d_in[6];
  const float* b2  = (const float*)d_in[7];
  const float* g2  = (const float*)d_in[8];
  const float* be2 = (const float*)d_in[9];
  const float* W3  = (const float*)d_in[10];
  const float* b3  = (const float*)d_in[11];

  const int N = in_sizes[0] / 64;
  const int E = in_sizes[1] / 2;
  const int* src = ei;       // edge_index[0]
  const int* dst = ei + E;   // edge_index[1]
  const size_t n64 = (size_t)N * 64;

  float* ws    = (float*)d_ws;
  float* dinv  = ws;                 // N
  float* hw    = dinv + N;           // N*64
  float* agg   = hw + n64;           // N*64
  float* h     = agg + n64;          // N*64
  float* stats = h + n64;            // 128
  float* coef  = stats + 128;        // 128

  const dim3 blk(256);
  const int total = E + N;
  const int scatterBlocks = (total + 7) / 8;          // 8 waves/block, 1 wave/edge
  const int gemmBlocks    = (N + 31) / 32;            // 32 rows/block
  const int ewBlocks      = (int)((n64 + 255) / 256);

  // degree -> dinv (in place)
  k_deg_init <<<(N + 255) / 256, blk, 0, stream>>>(dinv, N);
  k_deg_accum<<<(E + 255) / 256, blk, 0, stream>>>(dst, dinv, E);
  k_dinv     <<<(N + 255) / 256, blk, 0, stream>>>(dinv, N);

  auto layer = [&](const float* in, const float* W, const float* b,
                   const float* g, const float* be, float* out) {
    k_zero<<<2048, blk, 0, stream>>>(agg, n64);
    k_zero<<<1, blk, 0, stream>>>(stats, 128);
    k_gemm_n64<<<gemmBlocks, blk, 0, stream>>>(in, W, hw, N);
    k_scatter<<<scatterBlocks, blk, 0, stream>>>(src, dst, dinv, hw, agg, E, N);
    k_bn_stats<<<512, blk, 0, stream>>>(agg, b, stats, N);
    k_bn_finalize<<<1, 64, 0, stream>>>(stats, g, be, coef, 1.0f / (float)N);
    k_bn_apply_relu<<<ewBlocks, blk, 0, stream>>>(agg, b, coef, out, n64);
  };

  layer(x, W1, b1, g1, be1, h);   // layer 1: x -> h
  layer(h, W2, b2, g2, be2, h);   // layer 2: h -> h (hw consumed before overwrite)

  // layer 3: no BN/ReLU; out = agg + b3 + x (residual)
  k_zero<<<2048, blk, 0, stream>>>(agg, n64);
  k_gemm_n64<<<gemmBlocks, blk, 0, stream>>>(h, W3, hw, N);
  k_scatter<<<scatterBlocks, blk, 0, stream>>>(src, dst, dinv, hw, agg, E, N);
  k_final_add<<<ewBlocks, blk, 0, stream>>>(agg, b3, x, (float*)d_out, n64);
}